// GraphNetEncoder_14654428413991
// MI455X (gfx1250) — compile-verified
//
#include <hip/hip_runtime.h>

typedef float v2f __attribute__((ext_vector_type(2)));
typedef float v8f __attribute__((ext_vector_type(8)));

#define TILE_LD 68            // padded row stride (floats) -> conflict-free b64 LDS reads
#define TILE_SZ (16 * TILE_LD)

__device__ __forceinline__ v8f wmma4(v2f a, v2f b, v8f c) {
  // D(16x16,f32) += A(16x4,f32) * B(4x16,f32)
  return __builtin_amdgcn_wmma_f32_16x16x4_f32(false, a, false, b, (short)0, c, false, false);
}

// ---------------------------------------------------------------------------
// Generic fused (up to 2-layer) MLP over 64-wide hidden size.
//   h1 = [concat(s0..s3)] @ W1 + b1 + gA[gIdx[row]] + gB[row>>gShift]
//   h1 = relu1 ? relu(h1) : h1
//   if twoLayer: out = relu(h1 @ W2 + b2) * mask(row)
//   else:        out = h1
// Block = 128 threads = 4 waves; wave w handles rows [blk*64 + w*16, +16) x 64 cols.
// ---------------------------------------------------------------------------
__global__ __launch_bounds__(128) void mlp_kernel(
    const float* __restrict__ s0, int k0,
    const float* __restrict__ s1, int k1,
    const float* __restrict__ s2, int k2,
    const float* __restrict__ s3, int k3,
    const float* __restrict__ W1, const float* __restrict__ b1,
    const float* __restrict__ gA, const int* __restrict__ gIdx, int gClamp,
    const float* __restrict__ gB, int gShift,
    const int* __restrict__ maskIdx, int maskN,
    int twoLayer, int relu1,
    const float* __restrict__ W2, const float* __restrict__ b2,
    float* __restrict__ out, int M)
{
  __shared__ float lds[4 * TILE_SZ];
  const int wave = threadIdx.x >> 5;
  const int lane = threadIdx.x & 31;
  const int h    = lane >> 4;     // half-wave select
  const int m    = lane & 15;     // row (A) / col (B) within tile
  const int rowBase = blockIdx.x * 64 + wave * 16;
  float* t = lds + wave * TILE_SZ;

  // ---------------- layer 1: GEMM over concatenated segments ----------------
  v8f acc[4];
  #pragma unroll
  for (int c = 0; c < 4; c++) acc[c] = (v8f)0.0f;

  int arow = rowBase + m; if (arow > M - 1) arow = M - 1;

  const float* segp[4] = {s0, s1, s2, s3};
  const int    segk[4] = {k0, k1, k2, k3};
  int kbase = 0;
  for (int s = 0; s < 4; s++) {
    const float* sp = segp[s];
    const int kk = segk[s];
    if (!sp || kk <= 0) continue;
    const float* arp = sp + (size_t)arow * kk;
    for (int ks = 0; ks < kk; ks += 4) {
      v2f a = *(const v2f*)(arp + ks + 2 * h);          // A[m][k0+2h .. +1]
      const int kr = (kbase + ks + 2 * h) * 64;
      #pragma unroll
      for (int c = 0; c < 4; c++) {
        v2f b;
        b.x = W1[kr +       16 * c + m];                // B[k0+2h  ][16c+m]
        b.y = W1[kr + 64  + 16 * c + m];                // B[k0+2h+1][16c+m]
        acc[c] = wmma4(a, b, acc[c]);
      }
    }
    kbase += kk;
  }

  // store C fragments -> LDS tile (row r+8h, col 16c+m)
  #pragma unroll
  for (int c = 0; c < 4; c++)
    #pragma unroll
    for (int r = 0; r < 8; r++)
      t[(r + 8 * h) * TILE_LD + 16 * c + m] = acc[c][r];
  __syncthreads();

  // ---------------- bias / gather / relu pass (coalesced, float2/lane) ------
  {
    v2f bv = (v2f)0.0f;
    if (b1) bv = *(const v2f*)(b1 + 2 * lane);
    for (int rr = 0; rr < 16; rr++) {
      int row = rowBase + rr;
      int rc  = (row < M) ? row : (M - 1);
      v2f g = (v2f)0.0f;
      if (gA) {
        int gi = gIdx[rc];
        if (gi > gClamp - 1) gi = gClamp - 1;           // JAX clamp-gather semantics
        g += *(const v2f*)(gA + (size_t)gi * 64 + 2 * lane);
      }
      if (gB) g += *(const v2f*)(gB + (size_t)(rc >> gShift) * 64 + 2 * lane);
      v2f hv = *(const v2f*)(t + rr * TILE_LD + 2 * lane) + bv + g;
      if (relu1) { hv.x = fmaxf(hv.x, 0.0f); hv.y = fmaxf(hv.y, 0.0f); }
      if (!twoLayer) {
        if (row < M) *((v2f*)(out + (size_t)row * 64 + 2 * lane)) = hv;
      } else {
        *((v2f*)(t + rr * TILE_LD + 2 * lane)) = hv;
      }
    }
  }
  if (!twoLayer) return;
  __syncthreads();

  // ---------------- layer 2: h1(16x64) @ W2(64x64) ---------------------------
  #pragma unroll
  for (int c = 0; c < 4; c++) acc[c] = (v8f)0.0f;
  #pragma unroll
  for (int kk = 0; kk < 64; kk += 4) {
    v2f a = *(const v2f*)(t + m * TILE_LD + kk + 2 * h);
    const int kr = (kk + 2 * h) * 64;
    #pragma unroll
    for (int c = 0; c < 4; c++) {
      v2f b;
      b.x = W2[kr +      16 * c + m];
      b.y = W2[kr + 64 + 16 * c + m];
      acc[c] = wmma4(a, b, acc[c]);
    }
  }
  __syncthreads();
  #pragma unroll
  for (int c = 0; c < 4; c++)
    #pragma unroll
    for (int r = 0; r < 8; r++)
      t[(r + 8 * h) * TILE_LD + 16 * c + m] = acc[c][r];
  __syncthreads();

  // ---------------- bias2 / relu / mask / store ------------------------------
  {
    v2f b2v = (v2f)0.0f;
    if (b2) b2v = *(const v2f*)(b2 + 2 * lane);
    for (int rr = 0; rr < 16; rr++) {
      int row = rowBase + rr;
      if (row >= M) break;                               // uniform per wave
      v2f o = *(const v2f*)(t + rr * TILE_LD + 2 * lane) + b2v;
      o.x = fmaxf(o.x, 0.0f); o.y = fmaxf(o.y, 0.0f);
      if (maskIdx && (maskIdx[row] >= maskN)) o = (v2f)0.0f;
      *((v2f*)(out + (size_t)row * 64 + 2 * lane)) = o;
    }
  }
}

// ---------------------------------------------------------------------------
// Small helper kernels
// ---------------------------------------------------------------------------
__global__ void fill_kernel(float* p, int n) {
  int i = blockIdx.x * blockDim.x + threadIdx.x;
  if (i < n) p[i] = 0.0f;
}

// incoming[idx[f]] += e[f]  (segment_sum), 32 threads / edge, 2 cols / thread
__global__ void scatter_kernel(const float* __restrict__ e, const int* __restrict__ idx,
                               float* __restrict__ inc, int F, int N) {
  int tid = blockIdx.x * blockDim.x + threadIdx.x;
  int f = tid >> 5;
  int c = (tid & 31) * 2;
  if (f >= F) return;
  int gi = idx[f];
  if (gi >= N) return;                                   // masked edges contribute 0
  unsafeAtomicAdd(&inc[(size_t)gi * 64 + c],     e[(size_t)f * 64 + c]);
  unsafeAtomicAdd(&inc[(size_t)gi * 64 + c + 1], e[(size_t)f * 64 + c + 1]);
}

// outgoing[n] = sum_j new_edges[n*16+j]
__global__ void outgoing_kernel(const float* __restrict__ e, float* __restrict__ outg, int N) {
  int tid = blockIdx.x * blockDim.x + threadIdx.x;
  int n = tid >> 5;
  int c = (tid & 31) * 2;
  if (n >= N) return;
  const float* base = e + (size_t)n * 16 * 64 + c;
  float s0 = 0.0f, s1 = 0.0f;
  #pragma unroll
  for (int j = 0; j < 16; j++) { s0 += base[j * 64]; s1 += base[j * 64 + 1]; }
  outg[(size_t)n * 64 + c] = s0;
  outg[(size_t)n * 64 + c + 1] = s1;
}

// out[c] += column sum of x[M,64] (atomic; out must be zeroed first)
__global__ void colsum_kernel(const float* __restrict__ x, int M, float* __restrict__ out) {
  int c = threadIdx.x;                                   // 64 threads
  float s = 0.0f;
  for (int r = blockIdx.x; r < M; r += gridDim.x) s += x[(size_t)r * 64 + c];
  unsafeAtomicAdd(&out[c], s);
}

// out[c] = b[c] + [g0||g1](128) @ W[128,64]   (effective-bias fold of broadcast globals)
__global__ void tiny_lin_kernel(const float* __restrict__ b, const float* __restrict__ g0,
                                const float* __restrict__ g1, const float* __restrict__ W,
                                float* __restrict__ out) {
  int c = threadIdx.x;                                   // 64 threads
  float s = b[c];
  for (int k = 0; k < 64; k++) s += g0[k] * W[k * 64 + c];
  for (int k = 0; k < 64; k++) s += g1[k] * W[(64 + k) * 64 + c];
  out[c] = s;
}

// 2-layer MLP on a single row made of up to 4 segments
__global__ void tiny_mlp2_kernel(const float* s0, int k0, const float* s1, int k1,
                                 const float* s2, int k2, const float* s3, int k3,
                                 const float* __restrict__ W1, const float* __restrict__ b1,
                                 const float* __restrict__ W2, const float* __restrict__ b2,
                                 float* __restrict__ out) {
  __shared__ float hsh[64];
  int c = threadIdx.x;                                   // 64 threads
  float a = b1[c];
  int kb = 0;
  if (s0) { for (int k = 0; k < k0; k++) a += s0[k] * W1[(kb + k) * 64 + c]; kb += k0; }
  if (s1) { for (int k = 0; k < k1; k++) a += s1[k] * W1[(kb + k) * 64 + c]; kb += k1; }
  if (s2) { for (int k = 0; k < k2; k++) a += s2[k] * W1[(kb + k) * 64 + c]; kb += k2; }
  if (s3) { for (int k = 0; k < k3; k++) a += s3[k] * W1[(kb + k) * 64 + c]; kb += k3; }
  hsh[c] = fmaxf(a, 0.0f);
  __syncthreads();
  float o = b2[c];
  for (int k = 0; k < 64; k++) o += hsh[k] * W2[k * 64 + c];
  out[c] = fmaxf(o, 0.0f);
}

// ---------------------------------------------------------------------------
// Host-side orchestration
// ---------------------------------------------------------------------------
extern "C" void kernel_launch(void* const* d_in, const int* in_sizes, int n_in,
                              void* d_out, int out_size, void* d_ws, size_t ws_size,
                              hipStream_t stream) {
  const int N = 20000, F = 320000;                       // F = N * MAX_DEG

  const float* nodes = (const float*)d_in[0];            // [N,16]
  const float* edges = (const float*)d_in[1];            // [N,16,4] -> [F,4]
  const float* glb   = (const float*)d_in[2];            // [4]
  const int*   eidx  = (const int*)d_in[3];              // [F]

  // params leaves in jax tree order (dict keys sorted: enc_edge, enc_glob,
  // enc_node, rec; rec dicts sorted: edge, glob, node; each MLP: W0,b0,W1,b1)
  #define PW(i) ((const float*)d_in[i])
  const float *EeW0=PW(4),  *Eeb0=PW(5),  *EeW1=PW(6),  *Eeb1=PW(7);
  const float *EgW0=PW(8),  *Egb0=PW(9),  *EgW1=PW(10), *Egb1=PW(11);
  const float *EnW0=PW(12), *Enb0=PW(13), *EnW1=PW(14), *Enb1=PW(15);
  struct RecP { const float *eW0,*eb0,*eW1,*eb1,*gW0,*gb0,*gW1,*gb1,*nW0,*nb0,*nW1,*nb1; };
  RecP R[2];
  for (int r = 0; r < 2; r++) {
    int b = 16 + r * 12;
    R[r] = { PW(b+0),PW(b+1),PW(b+2),PW(b+3),      // edge MLP  (W1:[512,64])
             PW(b+4),PW(b+5),PW(b+6),PW(b+7),      // glob MLP  (W1:[256,64])
             PW(b+8),PW(b+9),PW(b+10),PW(b+11) };  // node MLP  (W1:[384,64])
  }
  #undef PW

  // workspace layout (floats)
  float* ws = (float*)d_ws;
  size_t oEN = 0;
  size_t oEE  = oEN  + (size_t)N * 64;
  size_t oEG  = oEE  + (size_t)F * 64;
  size_t oPIN = oEG  + 64;
  size_t oPOUT= oPIN + (size_t)N * 64;
  size_t oNE0 = oPOUT+ (size_t)N * 64;
  size_t oINC = oNE0 + (size_t)F * 64;
  size_t oOUT = oINC + (size_t)N * 64;
  size_t oNN0 = oOUT + (size_t)N * 64;
  size_t oG0  = oNN0 + (size_t)N * 64;
  size_t oEB1 = oG0  + 64;
  size_t oEB2 = oEB1 + 64;
  size_t oSN  = oEB2 + 64;
  size_t oSE  = oSN  + 64;
  size_t total= oSE  + 64;
  if (ws_size < total * sizeof(float)) return;

  float *en = ws+oEN, *ee = ws+oEE, *eg = ws+oEG, *Pin = ws+oPIN, *Pout = ws+oPOUT;
  float *ne0 = ws+oNE0, *inc = ws+oINC, *outg = ws+oOUT, *nn0 = ws+oNN0, *g0 = ws+oG0;
  float *eb1 = ws+oEB1, *eb2 = ws+oEB2, *sumN = ws+oSN, *sumE = ws+oSE;

  float* outAll = (float*)d_out;
  float* nOut = outAll;
  float* eOut = outAll + (size_t)N * 64;
  float* gOut = outAll + (size_t)N * 64 + (size_t)F * 64;

  auto launch_mlp = [&](const float* s0,int k0,const float* s1,int k1,
                        const float* s2,int k2,const float* s3,int k3,
                        const float* W1,const float* b1,
                        const float* gA,const int* gIdx,int gClamp,
                        const float* gB,int gShift,
                        const int* maskI,int maskN,int two,int relu1,
                        const float* W2,const float* b2,float* out,int M) {
    int grid = (M + 63) / 64;
    mlp_kernel<<<grid, 128, 0, stream>>>(s0,k0,s1,k1,s2,k2,s3,k3,W1,b1,
        gA,gIdx,gClamp,gB,gShift,maskI,maskN,two,relu1,W2,b2,out,M);
  };

  // ---------------- encoders ----------------
  launch_mlp(nodes,16, nullptr,0, nullptr,0, nullptr,0, EnW0,Enb0,
             nullptr,nullptr,0, nullptr,0, nullptr,0, 1,1, EnW1,Enb1, en, N);
  launch_mlp(edges,4, nullptr,0, nullptr,0, nullptr,0, EeW0,Eeb0,
             nullptr,nullptr,0, nullptr,0, nullptr,0, 1,1, EeW1,Eeb1, ee, F);
  tiny_mlp2_kernel<<<1,64,0,stream>>>(glb,4, nullptr,0, nullptr,0, nullptr,0,
                                      EgW0,Egb0,EgW1,Egb1, eg);

  // ---------------- recursions ----------------
  const float *nprev = en, *eprev = ee, *gprev = eg;
  for (int r = 0; r < 2; r++) {
    const RecP& p = R[r];
    float* oN = r ? nOut : nn0;
    float* oE = r ? eOut : ne0;
    float* oG = r ? gOut : g0;

    // P_in = [nprev||en] @ W1e[128:256],  P_out = [nprev||en] @ W1e[256:384]
    launch_mlp(nprev,64, en,64, nullptr,0, nullptr,0, p.eW0 + 128*64, nullptr,
               nullptr,nullptr,0, nullptr,0, nullptr,0, 0,0, nullptr,nullptr, Pin, N);
    launch_mlp(nprev,64, en,64, nullptr,0, nullptr,0, p.eW0 + 256*64, nullptr,
               nullptr,nullptr,0, nullptr,0, nullptr,0, 0,0, nullptr,nullptr, Pout, N);
    // effective bias: b1e + [gprev||eg] @ W1e[384:512]
    tiny_lin_kernel<<<1,64,0,stream>>>(p.eb0, gprev, eg, p.eW0 + 384*64, eb1);

    // edge MLP: relu( [eprev||ee]@We + Pin[idx] + Pout[row>>4] + eb1 ) -> layer2 -> *mask
    launch_mlp(eprev,64, ee,64, nullptr,0, nullptr,0, p.eW0, eb1,
               Pin, eidx, N, Pout, 4, eidx, N, 1,1, p.eW1, p.eb1, oE, F);

    // incoming = segment_sum(new_edges, edge_idx)
    fill_kernel<<<(N*64 + 255)/256, 256, 0, stream>>>(inc, N*64);
    scatter_kernel<<<(F*32 + 255)/256, 256, 0, stream>>>(oE, eidx, inc, F, N);
    // outgoing = new_edges.sum(axis=1)
    outgoing_kernel<<<(N*32 + 255)/256, 256, 0, stream>>>(oE, outg, N);

    // node MLP: effbias folds broadcast globals (W1n rows 256:384)
    tiny_lin_kernel<<<1,64,0,stream>>>(p.nb0, gprev, eg, p.nW0 + 256*64, eb2);
    launch_mlp(nprev,64, en,64, inc,64, outg,64, p.nW0, eb2,
               nullptr,nullptr,0, nullptr,0, nullptr,0, 1,1, p.nW1, p.nb1, oN, N);

    // global MLP: concat([new_nodes.sum(0), new_edges.sum((0,1)), gprev, eg])
    fill_kernel<<<1, 64, 0, stream>>>(sumN, 64);
    fill_kernel<<<1, 64, 0, stream>>>(sumE, 64);
    colsum_kernel<<<256, 64, 0, stream>>>(oN, N, sumN);
    colsum_kernel<<<256, 64, 0, stream>>>(oE, F, sumE);
    tiny_mlp2_kernel<<<1,64,0,stream>>>(sumN,64, sumE,64, gprev,64, eg,64,
                                        p.gW0, p.gb0, p.gW1, p.gb1, oG);

    nprev = oN; eprev = oE; gprev = oG;
  }
}